// _FTConv_82841329205582
// MI455X (gfx1250) — compile-verified
//
#include <hip/hip_runtime.h>
#include <hip/hip_bf16.h>

// ---------------------------------------------------------------------------
// Banded FFT-conv as three stages of dense DFT GEMMs on fp32 WMMA.
//   S  = signal (512x3000)  @ Ffwd (3000x1952)            [WMMA GEMM]
//   X  = per-freq complex mix of S with on-the-fly kernel DFT (band-masked)
//   out= X (1024x1952) @ Binv (1952x2976) + bias          [WMMA GEMM]
// All DFT bases generated on device with exact integer angle reduction.
// ---------------------------------------------------------------------------

typedef float v2f __attribute__((ext_vector_type(2)));
typedef float v8f __attribute__((ext_vector_type(8)));

#define LSIG   3000
#define NFREQ  973          // live frequencies f = 0..972
#define FPAD   976          // padded complex freq count (multiple of 16)
#define KF     (2*FPAD)     // 1952 real columns (re,im interleaved)
#define TOUT   2976         // L - K + 1
#define NB     8
#define CIN    64
#define COUT   128
#define KW     25
#define TWO_PI 6.28318530717958647692f

// ---- basis generation ------------------------------------------------------

__global__ void gen_fwd_basis(float* __restrict__ F) {
    int id = blockIdx.x * blockDim.x + threadIdx.x;
    if (id >= LSIG * FPAD) return;
    int n = id / FPAD;
    int f = id % FPAD;
    float c = 0.f, s = 0.f;
    if (f < NFREQ) {
        int p = (n * f) % LSIG;                 // exact reduction
        float th = (TWO_PI / (float)LSIG) * (float)p;
        __sincosf(th, &s, &c);
    }
    F[(size_t)n * KF + 2 * f]     = c;          // Re e^{-i th} contribution
    F[(size_t)n * KF + 2 * f + 1] = -s;         // Im e^{-i th}
}

__global__ void gen_inv_basis(float* __restrict__ Bi) {
    int id = blockIdx.x * blockDim.x + threadIdx.x;
    if (id >= FPAD * TOUT) return;
    int f = id / TOUT;
    int t = id % TOUT;
    float c = 0.f, s = 0.f;
    if (f < NFREQ) {
        int p = (f * t) % LSIG;
        float th = (TWO_PI / (float)LSIG) * (float)p;
        __sincosf(th, &s, &c);
        float a = (f == 0) ? (1.0f / (float)LSIG) : (2.0f / (float)LSIG);
        c *= a; s *= a;
    }
    Bi[(size_t)(2 * f)     * TOUT + t] = c;     // multiplies Re X[f]
    Bi[(size_t)(2 * f + 1) * TOUT + t] = -s;    // multiplies Im X[f]
}

__global__ void gen_twiddle(float* __restrict__ Ec, float* __restrict__ Es) {
    int id = blockIdx.x * blockDim.x + threadIdx.x;
    if (id >= NFREQ * KW) return;
    int f = id / KW;
    int k = id % KW;
    int p = (f * k) % LSIG;
    float th = (TWO_PI / (float)LSIG) * (float)p;
    float s, c;
    __sincosf(th, &s, &c);                      // e^{+i th} (conj of rfft)
    Ec[id] = c;
    Es[id] = s;
}

// ---- fp32 WMMA GEMM: C(MxN) = A(MxK) * B(KxN) [+ bias[row % biasmod]] ------
// One wave (32 threads) per 32x32 C tile; K stepped by 4 with
// V_WMMA_F32_16X16X4_F32. M,N multiples of 32, K multiple of 4.

__global__ __launch_bounds__(32)
void wmma_gemm_f32(const float* __restrict__ A, const float* __restrict__ Bm,
                   float* __restrict__ C, int M, int N, int K,
                   const float* __restrict__ bias, int biasmod) {
    const int lane = threadIdx.x & 31;
    const int n0 = blockIdx.x * 32;
    const int m0 = blockIdx.y * 32;
    const int lm = lane & 15;           // row within A fragment / col within B
    const int kh = (lane >> 4) * 2;     // K sub-offset held by this half-wave

    v8f acc00 = {}, acc01 = {}, acc10 = {}, acc11 = {};

    const float* Arow0 = A + (size_t)(m0 + lm) * K;
    const float* Arow1 = A + (size_t)(m0 + 16 + lm) * K;

    for (int k = 0; k < K; k += 4) {
        const int kk = k + kh;
        // A fragments (16x4), two M tiles; rows contiguous -> 8B vector loads
        v2f a0 = *(const v2f*)(Arow0 + kk);
        v2f a1 = *(const v2f*)(Arow1 + kk);
        // B fragments (4x16), two N tiles
        const float* Bp = Bm + (size_t)kk * N;
        v2f b0, b1;
        b0.x = Bp[n0 + lm];        b1.x = Bp[n0 + 16 + lm];
        b0.y = Bp[(size_t)N + n0 + lm];
        b1.y = Bp[(size_t)N + n0 + 16 + lm];

        acc00 = __builtin_amdgcn_wmma_f32_16x16x4_f32(false, a0, false, b0,
                                                      (short)0, acc00, false, false);
        acc01 = __builtin_amdgcn_wmma_f32_16x16x4_f32(false, a0, false, b1,
                                                      (short)0, acc01, false, false);
        acc10 = __builtin_amdgcn_wmma_f32_16x16x4_f32(false, a1, false, b0,
                                                      (short)0, acc10, false, false);
        acc11 = __builtin_amdgcn_wmma_f32_16x16x4_f32(false, a1, false, b1,
                                                      (short)0, acc11, false, false);
    }

    // C/D layout: VGPR r -> row r + 8*(lane>>4), col lane&15
    const int cr = (lane >> 4) * 8;
    const int cn = lane & 15;
#pragma unroll
    for (int r = 0; r < 8; ++r) {
        const int rA = m0 + r + cr;         // M tile 0
        const int rB = rA + 16;             // M tile 1
        const float bA = bias ? bias[rA % biasmod] : 0.0f;
        const float bB = bias ? bias[rB % biasmod] : 0.0f;
        C[(size_t)rA * N + n0 + cn]      = acc00[r] + bA;
        C[(size_t)rA * N + n0 + 16 + cn] = acc01[r] + bA;
        C[(size_t)rB * N + n0 + cn]      = acc10[r] + bB;
        C[(size_t)rB * N + n0 + 16 + cn] = acc11[r] + bB;
    }
}

// ---- per-frequency channel mixing (band masked) ----------------------------
// X[b*128+o][2f..2f+1] = sum_i S[b*64+i][f] * conj(rfft(w[o,i]))[f]  (f in band)
// kernel DFT built on the fly from the 973x25 twiddle table.

__global__ void band_combine(const float* __restrict__ S,
                             const float* __restrict__ W,
                             const float* __restrict__ Ec,
                             const float* __restrict__ Es,
                             float* __restrict__ X) {
    int id = blockIdx.x * blockDim.x + threadIdx.x;
    if (id >= COUT * FPAD) return;
    const int o = id / FPAD;
    const int f = id % FPAD;
    int band = o / 25; if (band > 4) band = 4;
    const int blo[5] = {0, 122, 244, 366, 488};
    const int bhi[5] = {122, 244, 366, 488, 972};   // inclusive

    float xre[NB], xim[NB];
#pragma unroll
    for (int b = 0; b < NB; ++b) { xre[b] = 0.f; xim[b] = 0.f; }

    if (f >= blo[band] && f <= bhi[band]) {
        const float* ec = Ec + (size_t)f * KW;
        const float* es = Es + (size_t)f * KW;
        for (int i = 0; i < CIN; ++i) {
            const float* wp = W + ((size_t)o * CIN + i) * KW;
            float kre = 0.f, kim = 0.f;
#pragma unroll
            for (int k = 0; k < KW; ++k) {
                const float w = wp[k];
                kre += w * ec[k];
                kim += w * es[k];
            }
#pragma unroll
            for (int b = 0; b < NB; ++b) {
                const float sre = S[((size_t)b * CIN + i) * KF + 2 * f];
                const float sim = S[((size_t)b * CIN + i) * KF + 2 * f + 1];
                xre[b] += sre * kre - sim * kim;
                xim[b] += sre * kim + sim * kre;
            }
        }
    }
#pragma unroll
    for (int b = 0; b < NB; ++b) {
        X[((size_t)b * COUT + o) * KF + 2 * f]     = xre[b];
        X[((size_t)b * COUT + o) * KF + 2 * f + 1] = xim[b];
    }
}

// ---------------------------------------------------------------------------

extern "C" void kernel_launch(void* const* d_in, const int* in_sizes, int n_in,
                              void* d_out, int out_size, void* d_ws, size_t ws_size,
                              hipStream_t stream) {
    (void)in_sizes; (void)n_in; (void)out_size; (void)ws_size;
    const float* signal = (const float*)d_in[0];   // (8,64,3000)
    const float* weight = (const float*)d_in[1];   // (128,64,25)
    const float* bias   = (const float*)d_in[2];   // (128,)
    // d_in[3..7] are the fb bases; regenerated on-device instead.

    char* ws = (char*)d_ws;
    auto carve = [&](size_t bytes) -> float* {
        float* p = (float*)ws;
        ws += (bytes + 255) & ~(size_t)255;
        return p;
    };
    float* Ffwd = carve((size_t)LSIG * KF * sizeof(float));   // 23.4 MB
    float* Binv = carve((size_t)KF * TOUT * sizeof(float));   // 23.2 MB
    float* S    = carve((size_t)NB * CIN * KF * sizeof(float));  // 4.0 MB
    float* X    = carve((size_t)NB * COUT * KF * sizeof(float)); // 8.0 MB
    float* Ec   = carve((size_t)NFREQ * KW * sizeof(float));
    float* Es   = carve((size_t)NFREQ * KW * sizeof(float));

    const int thr = 256;
    gen_fwd_basis<<<(LSIG * FPAD + thr - 1) / thr, thr, 0, stream>>>(Ffwd);
    gen_inv_basis<<<(FPAD * TOUT + thr - 1) / thr, thr, 0, stream>>>(Binv);
    gen_twiddle<<<(NFREQ * KW + thr - 1) / thr, thr, 0, stream>>>(Ec, Es);

    // S = signal @ Ffwd : M=512, N=1952, K=3000
    dim3 g1(KF / 32, (NB * CIN) / 32);
    wmma_gemm_f32<<<g1, 32, 0, stream>>>(signal, Ffwd, S,
                                         NB * CIN, KF, LSIG, nullptr, 1);

    band_combine<<<(COUT * FPAD + thr - 1) / thr, thr, 0, stream>>>(S, weight, Ec, Es, X);

    // out = X @ Binv + bias : M=1024, N=2976, K=1952
    dim3 g2(TOUT / 32, (NB * COUT) / 32);
    wmma_gemm_f32<<<g2, 32, 0, stream>>>(X, Binv, (float*)d_out,
                                         NB * COUT, TOUT, KF, bias, COUT);
}